// EuclideanUniverse_79439715107371
// MI455X (gfx1250) — compile-verified
//
#include <hip/hip_runtime.h>
#include <math.h>

#define NPT 4096
#define CJ 4
#define JCH (NPT / CJ) /* 1024 */

#define D_MIN 0.08f
#define R_H 5.0f
#define DT_C 0.01f
#define KDT_C 0.05f
#define TWO_PI_F 6.28318530717958647692f

typedef __attribute__((ext_vector_type(2))) float v2f;
typedef __attribute__((ext_vector_type(8))) float v8f;

__device__ __forceinline__ float fsqrt_fast(float x) { return __builtin_amdgcn_sqrtf(x); }
__device__ __forceinline__ float frcp_fast(float x) { return __builtin_amdgcn_rcpf(x); }
__device__ __forceinline__ float sigm(float x) { return frcp_fast(1.0f + __expf(-x)); }

// ---------------- K0: norms + sin/cos of theta ----------------
__global__ __launch_bounds__(256) void k_prep(const float* __restrict__ h,
                                              const float* __restrict__ th,
                                              float* n0, float* s0, float* c0) {
  int i = blockIdx.x * 256 + threadIdx.x;
  float4 hv = *(const float4*)(h + i * 4);
  n0[i] = hv.x * hv.x + hv.y * hv.y + hv.z * hv.z + hv.w * hv.w;
  float t = th[i];
  s0[i] = __sinf(t);
  c0[i] = __cosf(t);
}

// ---------------- K1: pass A — forces + Kuramoto f1 sums (WMMA Gram) ----------------
__global__ __launch_bounds__(128) void k_passA(const float* __restrict__ h,
                                               const float* __restrict__ n0,
                                               const float* __restrict__ s0,
                                               const float* __restrict__ c0,
                                               float* __restrict__ pA) {
  __shared__ __align__(16) float lh[JCH * 4];
  __shared__ float ln[JCH], ls[JCH], lc[JCH];
  const int tid = threadIdx.x;
  const int jc0 = blockIdx.y * JCH;
  for (int idx = tid; idx < JCH; idx += 128) {
    *(float4*)(&lh[idx * 4]) = *(const float4*)(h + (size_t)(jc0 + idx) * 4);
    ln[idx] = n0[jc0 + idx];
    ls[idx] = s0[jc0 + idx];
    lc[idx] = c0[jc0 + idx];
  }
  __syncthreads();
  const int wave = tid >> 5, lane = tid & 31;
  const int nl = lane & 15, hi = lane >> 4;
  const int i0 = (blockIdx.x * 4 + wave) * 16;
  // A tile: 16x4 f32, lane = M (%16), VGPR0/1 = K {2*hi, 2*hi+1}
  v2f a = *(const v2f*)(h + (size_t)(i0 + nl) * 4 + 2 * hi);
  float ni[8];
#pragma unroll
  for (int v = 0; v < 8; ++v) ni[v] = n0[i0 + v + 8 * hi];
  float aFx[8], aFy[8], aFz[8], aFw[8], aW[8], aS[8], aC[8];
#pragma unroll
  for (int v = 0; v < 8; ++v) { aFx[v]=aFy[v]=aFz[v]=aFw[v]=aW[v]=aS[v]=aC[v]=0.0f; }

  for (int jt = 0; jt < JCH; jt += 16) {
    const int lj = jt + nl;
    const int jg = jc0 + lj;
    float4 hj = *(const float4*)(&lh[lj * 4]);
    // B = H^T tile: b is a subset of hj (K = 2*hi, 2*hi+1)
    v2f b;
    b.x = hi ? hj.z : hj.x;
    b.y = hi ? hj.w : hj.y;
    v8f cz = {};
    v8f g = __builtin_amdgcn_wmma_f32_16x16x4_f32(false, a, false, b, (short)0, cz, false, false);
    float nj = ln[lj], sj = ls[lj], cj = lc[lj];
#pragma unroll
    for (int v = 0; v < 8; ++v) {
      int ig = i0 + v + 8 * hi;
      float off = (ig == jg) ? 0.0f : 1.0f;
      float sq = fmaxf(ni[v] + nj - 2.0f * g[v], 0.0f);
      float D = fsqrt_fast(sq);
      float D0 = (D < D_MIN) ? D + D_MIN : D;
      float De = fmaxf(D0, 0.05f);
      float dV = 2.0f * __expf(-2.0f * De) - 0.5f * __expf(-0.5f * De);
      float Kg = -dV * sigm(2.0f * (R_H - De));
      float Kk = off * 0.5f * __expf(-0.5f * D0);
      float dn = fmaxf(fsqrt_fast(fmaxf(sq, 1e-16f)), 1e-8f);
      float w = off * Kg * frcp_fast(dn);
      aFx[v] += w * hj.x; aFy[v] += w * hj.y; aFz[v] += w * hj.z; aFw[v] += w * hj.w;
      aW[v] += w; aS[v] += Kk * sj; aC[v] += Kk * cj;
    }
  }
  // reduce over the 16 lanes (N dim) within each half-wave
#pragma unroll
  for (int v = 0; v < 8; ++v) {
    for (int m = 1; m < 16; m <<= 1) {
      aFx[v] += __shfl_xor(aFx[v], m); aFy[v] += __shfl_xor(aFy[v], m);
      aFz[v] += __shfl_xor(aFz[v], m); aFw[v] += __shfl_xor(aFw[v], m);
      aW[v] += __shfl_xor(aW[v], m);  aS[v] += __shfl_xor(aS[v], m);
      aC[v] += __shfl_xor(aC[v], m);
    }
  }
  if (nl == 0) {
    float* base = pA + (size_t)blockIdx.y * 7 * NPT;
#pragma unroll
    for (int v = 0; v < 8; ++v) {
      int ig = i0 + v + 8 * hi;
      base[0 * NPT + ig] = aFx[v]; base[1 * NPT + ig] = aFy[v];
      base[2 * NPT + ig] = aFz[v]; base[3 * NPT + ig] = aFw[v];
      base[4 * NPT + ig] = aW[v];  base[5 * NPT + ig] = aS[v];
      base[6 * NPT + ig] = aC[v];
    }
  }
}

// ---------------- K2: finalize A — h_new, t_mid ----------------
__global__ __launch_bounds__(256) void k_finA(const float* __restrict__ h,
                                              const float* __restrict__ th,
                                              const float* __restrict__ om,
                                              const float* __restrict__ s0,
                                              const float* __restrict__ c0,
                                              const float* __restrict__ pA,
                                              float* __restrict__ hn, float* n1, float* rr,
                                              float* tmid, float* sm, float* cm) {
  int i = blockIdx.x * 256 + threadIdx.x;
  float Fx = 0, Fy = 0, Fz = 0, Fw = 0, SW = 0, S1 = 0, C1 = 0;
  for (int c = 0; c < CJ; ++c) {
    const float* b = pA + (size_t)c * 7 * NPT;
    Fx += b[0 * NPT + i]; Fy += b[1 * NPT + i]; Fz += b[2 * NPT + i]; Fw += b[3 * NPT + i];
    SW += b[4 * NPT + i]; S1 += b[5 * NPT + i]; C1 += b[6 * NPT + i];
  }
  float4 hv = *(const float4*)(h + (size_t)i * 4);
  Fx -= SW * hv.x; Fy -= SW * hv.y; Fz -= SW * hv.z; Fw -= SW * hv.w;
  float fn = sqrtf(fmaxf(Fx * Fx + Fy * Fy + Fz * Fz + Fw * Fw, 1e-16f));
  float s1 = fminf(6.0f / fmaxf(fn, 1e-8f), 1.0f);
  float dx = DT_C * Fx * s1, dy = DT_C * Fy * s1, dz = DT_C * Fz * s1, dw = DT_C * Fw * s1;
  float dn = sqrtf(fmaxf(dx * dx + dy * dy + dz * dz + dw * dw, 1e-16f));
  float s2 = fminf(0.5f / fmaxf(dn, 1e-8f), 1.0f);
  float nx = hv.x + dx * s2, ny = hv.y + dy * s2, nz = hv.z + dz * s2, nw = hv.w + dw * s2;
  if (nx != nx || ny != ny || nz != nz || nw != nw) { nx = hv.x; ny = hv.y; nz = hv.z; nw = hv.w; }
  hn[i * 4 + 0] = nx; hn[i * 4 + 1] = ny; hn[i * 4 + 2] = nz; hn[i * 4 + 3] = nw;
  float nn = nx * nx + ny * ny + nz * nz + nw * nw;
  n1[i] = nn; rr[i] = sqrtf(nn + 1e-16f);
  float f1 = om[i] + c0[i] * S1 - s0[i] * C1;
  float tm = fmodf(th[i] + 0.5f * KDT_C * f1, TWO_PI_F);
  if (tm < 0.0f) tm += TWO_PI_F;
  tmid[i] = tm; sm[i] = __sinf(tm); cm[i] = __cosf(tm);
}

// ---------------- K3: pass B — Kuramoto f2 sums (WMMA Gram over pre-step h) ----------------
__global__ __launch_bounds__(128) void k_passB(const float* __restrict__ h,
                                               const float* __restrict__ n0,
                                               const float* __restrict__ sm,
                                               const float* __restrict__ cm,
                                               float* __restrict__ pB) {
  __shared__ __align__(16) float lh[JCH * 4];
  __shared__ float ln[JCH], ls[JCH], lc[JCH];
  const int tid = threadIdx.x;
  const int jc0 = blockIdx.y * JCH;
  for (int idx = tid; idx < JCH; idx += 128) {
    *(float4*)(&lh[idx * 4]) = *(const float4*)(h + (size_t)(jc0 + idx) * 4);
    ln[idx] = n0[jc0 + idx]; ls[idx] = sm[jc0 + idx]; lc[idx] = cm[jc0 + idx];
  }
  __syncthreads();
  const int wave = tid >> 5, lane = tid & 31;
  const int nl = lane & 15, hi = lane >> 4;
  const int i0 = (blockIdx.x * 4 + wave) * 16;
  v2f a = *(const v2f*)(h + (size_t)(i0 + nl) * 4 + 2 * hi);
  float ni[8];
#pragma unroll
  for (int v = 0; v < 8; ++v) ni[v] = n0[i0 + v + 8 * hi];
  float aS[8], aC[8];
#pragma unroll
  for (int v = 0; v < 8; ++v) { aS[v] = aC[v] = 0.0f; }
  for (int jt = 0; jt < JCH; jt += 16) {
    const int lj = jt + nl;
    const int jg = jc0 + lj;
    float4 hj = *(const float4*)(&lh[lj * 4]);
    v2f b;
    b.x = hi ? hj.z : hj.x;
    b.y = hi ? hj.w : hj.y;
    v8f cz = {};
    v8f g = __builtin_amdgcn_wmma_f32_16x16x4_f32(false, a, false, b, (short)0, cz, false, false);
    float nj = ln[lj], sj = ls[lj], cj = lc[lj];
#pragma unroll
    for (int v = 0; v < 8; ++v) {
      int ig = i0 + v + 8 * hi;
      float off = (ig == jg) ? 0.0f : 1.0f;
      float sq = fmaxf(ni[v] + nj - 2.0f * g[v], 0.0f);
      float D = fsqrt_fast(sq);
      float D0 = (D < D_MIN) ? D + D_MIN : D;
      float Kk = off * 0.5f * __expf(-0.5f * D0);
      aS[v] += Kk * sj; aC[v] += Kk * cj;
    }
  }
#pragma unroll
  for (int v = 0; v < 8; ++v) {
    for (int m = 1; m < 16; m <<= 1) {
      aS[v] += __shfl_xor(aS[v], m);
      aC[v] += __shfl_xor(aC[v], m);
    }
  }
  if (nl == 0) {
    float* base = pB + (size_t)blockIdx.y * 2 * NPT;
#pragma unroll
    for (int v = 0; v < 8; ++v) {
      int ig = i0 + v + 8 * hi;
      base[0 * NPT + ig] = aS[v];
      base[1 * NPT + ig] = aC[v];
    }
  }
}

// ---------------- K4: finalize B — theta_new ----------------
__global__ __launch_bounds__(256) void k_finB(const float* __restrict__ th,
                                              const float* __restrict__ om,
                                              const float* __restrict__ sm,
                                              const float* __restrict__ cm,
                                              const float* __restrict__ pB,
                                              float* __restrict__ tout) {
  int i = blockIdx.x * 256 + threadIdx.x;
  float S2 = 0, C2 = 0;
  for (int c = 0; c < CJ; ++c) {
    const float* b = pB + (size_t)c * 2 * NPT;
    S2 += b[0 * NPT + i]; C2 += b[1 * NPT + i];
  }
  float f2 = om[i] + cm[i] * S2 - sm[i] * C2;
  float tn = fmodf(th[i] + KDT_C * f2, TWO_PI_F);
  if (tn < 0.0f) tn += TWO_PI_F;
  tout[i] = tn;
}

// ---------------- K5: pass C — E, sum(D1), causal, per-row 8 smallest ----------------
__global__ __launch_bounds__(256) void k_passC(const float* __restrict__ h,
                                               const float* __restrict__ hn,
                                               float* __restrict__ srow,
                                               float* __restrict__ p5) {
  __shared__ __align__(16) float l0[JCH * 4];
  __shared__ __align__(16) float l1[JCH * 4];
  __shared__ float red[256];
  const int tid = threadIdx.x;
  const int i = blockIdx.x * 256 + tid;
  float4 a0 = *(const float4*)(h + (size_t)i * 4);
  float4 a1 = *(const float4*)(hn + (size_t)i * 4);
  float m[8];
#pragma unroll
  for (int t = 0; t < 8; ++t) m[t] = 3.4e38f;
  float accE = 0, accD = 0, accZ = 0;
  for (int cch = 0; cch < CJ; ++cch) {
    __syncthreads();
    for (int idx = tid; idx < JCH; idx += 256) {
      *(float4*)(&l0[idx * 4]) = *(const float4*)(h + (size_t)(cch * JCH + idx) * 4);
      *(float4*)(&l1[idx * 4]) = *(const float4*)(hn + (size_t)(cch * JCH + idx) * 4);
    }
    __syncthreads();
    for (int lj = 0; lj < JCH; ++lj) {
      int j = cch * JCH + lj;
      if (j == i) continue;
      float4 b0 = *(const float4*)(&l0[lj * 4]);
      float4 b1 = *(const float4*)(&l1[lj * 4]);
      float dx = a0.x - b0.x, dy = a0.y - b0.y, dz = a0.z - b0.z, dw = a0.w - b0.w;
      float sq0 = dx * dx + dy * dy + dz * dz + dw * dw;
      float ex = a1.x - b1.x, ey = a1.y - b1.y, ez = a1.z - b1.z, ew = a1.w - b1.w;
      float sq1 = ex * ex + ey * ey + ez * ez + ew * ew;
      float D0r = fsqrt_fast(sq0);
      float D0v = (D0r < D_MIN) ? D0r + D_MIN : D0r;
      float De = fmaxf(D0v, 0.05f);
      float dV = 2.0f * __expf(-2.0f * De) - 0.5f * __expf(-0.5f * De);
      float Kga = fabsf(dV) * sigm(2.0f * (R_H - De));
      float D1r = fsqrt_fast(sq1);
      float D1v = (D1r < D_MIN) ? D1r + D_MIN : D1r;
      accE += __expf(-0.5f * D1v) - __expf(-2.0f * D1v);
      accD += D1v;
      if (D1v > R_H) accZ += Kga;
      if (D1v < m[7]) {
        m[7] = D1v;
#pragma unroll
        for (int t = 7; t > 0; --t) {
          if (m[t] < m[t - 1]) { float w = m[t]; m[t] = m[t - 1]; m[t - 1] = w; }
        }
      }
    }
  }
  srow[i] = ((m[0] + m[1]) + (m[2] + m[3])) + ((m[4] + m[5]) + (m[6] + m[7]));
  // block reductions (fixed order)
  float vals[3] = {accE, accD, accZ};
  for (int q = 0; q < 3; ++q) {
    __syncthreads();
    red[tid] = vals[q];
    __syncthreads();
    for (int st = 128; st > 0; st >>= 1) {
      if (tid < st) red[tid] += red[tid + st];
      __syncthreads();
    }
    if (tid == 0) p5[blockIdx.x * 3 + q] = red[0];
  }
}

// ---------------- K6a: global scalar reductions ----------------
__global__ __launch_bounds__(256) void k_red1(const float* __restrict__ srow,
                                              const float* __restrict__ rr,
                                              const float* __restrict__ p5,
                                              float* __restrict__ scal) {
  __shared__ float rs[256], rm[256];
  int tid = threadIdx.x;
  float s = 0, mx = -1e30f;
  for (int i = tid; i < NPT; i += 256) { s += srow[i]; mx = fmaxf(mx, rr[i]); }
  rs[tid] = s; rm[tid] = mx;
  __syncthreads();
  for (int st = 128; st > 0; st >>= 1) {
    if (tid < st) { rs[tid] += rs[tid + st]; rm[tid] = fmaxf(rm[tid], rm[tid + st]); }
    __syncthreads();
  }
  if (tid == 0) {
    float E = 0, Ds = 0, Cz = 0;
    for (int b = 0; b < 16; ++b) { E += p5[b * 3 + 0]; Ds += p5[b * 3 + 1]; Cz += p5[b * 3 + 2]; }
    scal[0] = E; scal[1] = Ds; scal[2] = Cz; scal[3] = rs[0]; scal[4] = rm[0];
    scal[5] = 0.8f * Ds / ((float)NPT * (float)NPT);  // eps = mean(D1)*0.8
  }
}

// ---------------- K7: pass D — adjacency sum (WMMA Gram over h_new) ----------------
__global__ __launch_bounds__(128) void k_passD(const float* __restrict__ hn,
                                               const float* __restrict__ n1,
                                               const float* __restrict__ scal,
                                               float* __restrict__ p7) {
  __shared__ __align__(16) float lh[JCH * 4];
  __shared__ float ln[JCH];
  __shared__ float red[128];
  const int tid = threadIdx.x;
  const int jc0 = blockIdx.y * JCH;
  for (int idx = tid; idx < JCH; idx += 128) {
    *(float4*)(&lh[idx * 4]) = *(const float4*)(hn + (size_t)(jc0 + idx) * 4);
    ln[idx] = n1[jc0 + idx];
  }
  __syncthreads();
  const float eps = scal[5];
  const int wave = tid >> 5, lane = tid & 31;
  const int nl = lane & 15, hi = lane >> 4;
  const int i0 = (blockIdx.x * 4 + wave) * 16;
  v2f a = *(const v2f*)(hn + (size_t)(i0 + nl) * 4 + 2 * hi);
  float ni[8];
#pragma unroll
  for (int v = 0; v < 8; ++v) ni[v] = n1[i0 + v + 8 * hi];
  float acc = 0;
  for (int jt = 0; jt < JCH; jt += 16) {
    const int lj = jt + nl;
    const int jg = jc0 + lj;
    float4 hj = *(const float4*)(&lh[lj * 4]);
    v2f b;
    b.x = hi ? hj.z : hj.x;
    b.y = hi ? hj.w : hj.y;
    v8f cz = {};
    v8f g = __builtin_amdgcn_wmma_f32_16x16x4_f32(false, a, false, b, (short)0, cz, false, false);
    float nj = ln[lj];
#pragma unroll
    for (int v = 0; v < 8; ++v) {
      int ig = i0 + v + 8 * hi;
      float off = (ig == jg) ? 0.0f : 1.0f;
      float sq = fmaxf(ni[v] + nj - 2.0f * g[v], 0.0f);
      float D = fsqrt_fast(sq);
      float D1 = (D < D_MIN) ? D + D_MIN : D;
      acc += off * frcp_fast(1.0f + __expf((D1 - eps) * (1.0f / 0.3f)));
    }
  }
  red[tid] = acc;
  __syncthreads();
  for (int st = 64; st > 0; st >>= 1) {
    if (tid < st) red[tid] += red[tid + st];
    __syncthreads();
  }
  if (tid == 0) p7[blockIdx.y * (NPT / 64) + blockIdx.x] = red[0];
}

// ---------------- K6b: radial entropy bins ----------------
__global__ __launch_bounds__(256) void k_bins(const float* __restrict__ rr,
                                              const float* __restrict__ scal,
                                              float* __restrict__ bins) {
  __shared__ float sb[256 * 8];
  int tid = threadIdx.x;
  int i = blockIdx.x * 256 + tid;
  float rmax = scal[4];
  float rn = rr[i] / (rmax + 1e-6f);
  float e[8], ss = 0;
#pragma unroll
  for (int b = 0; b < 8; ++b) {
    float cb = 0.0625f + 0.125f * (float)b;
    e[b] = __expf(-fabsf(rn - cb) * 20.0f);
    ss += e[b];
  }
  float inv = 1.0f / ss;
#pragma unroll
  for (int b = 0; b < 8; ++b) sb[tid * 8 + b] = e[b] * inv;
  __syncthreads();
  for (int st = 128; st > 0; st >>= 1) {
    if (tid < st) {
#pragma unroll
      for (int b = 0; b < 8; ++b) sb[tid * 8 + b] += sb[(tid + st) * 8 + b];
    }
    __syncthreads();
  }
  if (tid == 0) {
#pragma unroll
    for (int b = 0; b < 8; ++b) bins[blockIdx.x * 8 + b] = sb[b];
  }
}

// ---------------- K8: final loss combine ----------------
__global__ void k_final(const float* __restrict__ scal, const float* __restrict__ p7,
                        const float* __restrict__ bins, const float* __restrict__ Eref,
                        float* __restrict__ out) {
  if (threadIdx.x != 0 || blockIdx.x != 0) return;
  float adj = 0;
  for (int k = 0; k < 4 * (NPT / 64); ++k) adj += p7[k];
  float Ecur = 0.5f * scal[0];
  float er = Eref[0];
  float Le = (Ecur - er) / (fabsf(er) + 1e-6f);
  Le *= Le;
  float Lc = scal[3] / ((float)NPT * 8.0f);
  float topo = adj * 0.5f - (float)(NPT - 1);
  float Lt = (topo > 0.0f ? topo : 0.0f) / (float)NPT;
  float Lh = 0;
  for (int b = 0; b < 8; ++b) {
    float p = 0;
    for (int k = 0; k < 16; ++k) p += bins[k * 8 + b];
    p /= (float)NPT;
    p = fmaxf(p, 1e-8f);
    Lh -= p * __logf(p);
  }
  float Lz = scal[2] / ((float)NPT * (float)NPT);
  out[NPT * 5] = Le + Lc + Lt - 0.1f * Lh + Lz;
}

extern "C" void kernel_launch(void* const* d_in, const int* in_sizes, int n_in,
                              void* d_out, int out_size, void* d_ws, size_t ws_size,
                              hipStream_t stream) {
  (void)in_sizes; (void)n_in; (void)out_size; (void)ws_size;
  const float* h = (const float*)d_in[0];
  const float* th = (const float*)d_in[1];
  const float* om = (const float*)d_in[2];
  const float* er = (const float*)d_in[3];
  float* out = (float*)d_out;

  float* W = (float*)d_ws;
  float* n0 = W;                 // N
  float* s0 = W + 1 * NPT;       // N
  float* c0 = W + 2 * NPT;       // N
  float* tmid = W + 3 * NPT;     // N
  float* sm = W + 4 * NPT;       // N
  float* cm = W + 5 * NPT;       // N
  float* n1 = W + 6 * NPT;       // N
  float* rr = W + 7 * NPT;       // N
  float* srow = W + 8 * NPT;     // N
  float* pA = W + 9 * NPT;       // 7*CJ*N
  float* pB = pA + 7 * CJ * NPT; // 2*CJ*N
  float* p5 = pB + 2 * CJ * NPT; // 48 (pad 64)
  float* scal = p5 + 64;         // 16
  float* p7 = scal + 16;         // 256
  float* bins = p7 + 256;        // 128
  // total ws: ~740 KB of floats

  float* hn = out;               // h_new lives in d_out[0 .. 4N)
  float* tout = out + 4 * NPT;   // theta_new in d_out[4N .. 5N)

  k_prep<<<NPT / 256, 256, 0, stream>>>(h, th, n0, s0, c0);
  k_passA<<<dim3(NPT / 64, CJ), 128, 0, stream>>>(h, n0, s0, c0, pA);
  k_finA<<<NPT / 256, 256, 0, stream>>>(h, th, om, s0, c0, pA, hn, n1, rr, tmid, sm, cm);
  k_passB<<<dim3(NPT / 64, CJ), 128, 0, stream>>>(h, n0, sm, cm, pB);
  k_finB<<<NPT / 256, 256, 0, stream>>>(th, om, sm, cm, pB, tout);
  k_passC<<<NPT / 256, 256, 0, stream>>>(h, hn, srow, p5);
  k_red1<<<1, 256, 0, stream>>>(srow, rr, p5, scal);
  k_passD<<<dim3(NPT / 64, CJ), 128, 0, stream>>>(hn, n1, scal, p7);
  k_bins<<<NPT / 256, 256, 0, stream>>>(rr, scal, bins);
  k_final<<<1, 1, 0, stream>>>(scal, p7, bins, er, out);
}